// GATCN_64579128263347
// MI455X (gfx1250) — compile-verified
//
#include <hip/hip_runtime.h>
#include <hip/hip_bf16.h>

// ---------------------------------------------------------------------------
// 3-layer GAT (edge-featured) for MI455X / gfx1250, wave32.
// GEMMs use V_WMMA_F32_16X16X4_F32. Softmax aggregation uses f32/u32 atomics.
// ---------------------------------------------------------------------------

typedef float v2f __attribute__((ext_vector_type(2)));
typedef float v8f __attribute__((ext_vector_type(8)));

#define HF 128   // feature width (all layers)

__device__ __forceinline__ unsigned ordkey(float f) {
    unsigned u = __float_as_uint(f);
    return (u & 0x80000000u) ? ~u : (u | 0x80000000u);
}
__device__ __forceinline__ float deord(unsigned u) {
    u = (u & 0x80000000u) ? (u & 0x7fffffffu) : ~u;
    return __uint_as_float(u);
}
__device__ __forceinline__ float leaky(float x, float s) { return x > 0.f ? x : s * x; }

// --------------------------------------------------------------------------
// small helpers
// --------------------------------------------------------------------------
__global__ void k_zero(float* p, int n) {
    int i = blockIdx.x * blockDim.x + threadIdx.x;
    if (i < n) p[i] = 0.f;
}

// sum of edge_attr columns (E x 3) -> easum[3]
__global__ void k_ea_sum(const float* __restrict__ ea, int E, float* __restrict__ easum) {
    int idx = blockIdx.x * blockDim.x + threadIdx.x;
    int stride = gridDim.x * blockDim.x;
    float s0 = 0.f, s1 = 0.f, s2 = 0.f;
    for (int e = idx; e < E; e += stride) {
        s0 += ea[e * 3 + 0];
        s1 += ea[e * 3 + 1];
        s2 += ea[e * 3 + 2];
    }
    atomicAdd(easum + 0, s0);
    atomicAdd(easum + 1, s1);
    atomicAdd(easum + 2, s2);
}

// c = We @ a_edge (3-vector), cvec[3] = mean(ea) . c
__global__ void k_compute_c(const float* __restrict__ We, const float* __restrict__ ae,
                            const float* __restrict__ easum, float* __restrict__ cvec, int E) {
    if (threadIdx.x != 0 || blockIdx.x != 0) return;
    float cm = 0.f;
    for (int d = 0; d < 3; ++d) {
        float c = 0.f;
        for (int h = 0; h < HF; ++h) c += We[d * HF + h] * ae[h];
        cvec[d] = c;
        cm += (easum[d] / (float)E) * c;
    }
    cvec[3] = cm;
}

// --------------------------------------------------------------------------
// GEMM: Ho[N x 128] = X[N x 128] @ W[128 x 128], f32 WMMA 16x16x4.
// 8 waves/block; wave w owns a fixed 16-col tile (B panel held in 64 VGPRs),
// block grid-strides over 16-row strips.  A-rows beyond N are clamped (their
// D-rows are never stored), so the hot loop has no EXEC manipulation: the
// whole A panel is loaded in one unconditional burst, then 32 chained WMMAs.
// --------------------------------------------------------------------------
__global__ __launch_bounds__(256) void k_gemm_xw(const float* __restrict__ X,
                                                 const float* __restrict__ W,
                                                 float* __restrict__ Ho, int n) {
    const int lane = threadIdx.x & 31;
    const int wv   = threadIdx.x >> 5;   // 0..7 -> column tile
    const int lo16 = lane & 15;
    const int hi   = lane >> 4;          // 0/1
    const int ncol = wv * 16 + lo16;

    // Preload B: lane holds W[k][ncol] for k = 2*hi + 4*kk + {0,1}
    v2f breg[32];
#pragma unroll
    for (int kk = 0; kk < 32; ++kk) {
        int kb = kk * 4 + hi * 2;
        v2f b;
        b.x = W[(size_t)kb * HF + ncol];
        b.y = W[(size_t)(kb + 1) * HF + ncol];
        breg[kk] = b;
    }

    const int strips = (n + 15) >> 4;
    for (int s = blockIdx.x; s < strips; s += gridDim.x) {
        const int r0 = s * 16;
        int rowA = r0 + lo16;
        if (rowA >= n) rowA = n - 1;              // clamp: junk rows never stored
        const float* xrow = X + (size_t)rowA * HF + hi * 2;

        // phase 1: load the whole A panel (32 x b64 per lane, no branches)
        v2f a[32];
#pragma unroll
        for (int kk = 0; kk < 32; ++kk) {
            float2 t = *(const float2*)(xrow + kk * 4);
            a[kk].x = t.x;
            a[kk].y = t.y;
        }

        // phase 2: 32 chained WMMAs
        v8f acc = {0.f, 0.f, 0.f, 0.f, 0.f, 0.f, 0.f, 0.f};
#pragma unroll
        for (int kk = 0; kk < 32; ++kk) {
            acc = __builtin_amdgcn_wmma_f32_16x16x4_f32(
                false, a[kk], false, breg[kk], (short)0, acc, false, false);
        }

#pragma unroll
        for (int v = 0; v < 8; ++v) {
            int row = r0 + v + hi * 8;
            if (row < n) Ho[(size_t)row * HF + ncol] = acc[v];
        }
    }
}

// --------------------------------------------------------------------------
// als[i] = h[i].a_src ; ald[i] = h[i].a_dst  (one wave per node)
// --------------------------------------------------------------------------
__global__ __launch_bounds__(256) void k_node_dots(const float* __restrict__ Hm,
                                                   const float* __restrict__ avs,
                                                   const float* __restrict__ avd,
                                                   float* __restrict__ als,
                                                   float* __restrict__ ald, int n) {
    const int lane = threadIdx.x & 31;
    int wid = (blockIdx.x * blockDim.x + threadIdx.x) >> 5;
    int nw  = (gridDim.x * blockDim.x) >> 5;
    float s0 = avs[lane], s1 = avs[lane + 32], s2 = avs[lane + 64], s3 = avs[lane + 96];
    float d0 = avd[lane], d1 = avd[lane + 32], d2 = avd[lane + 64], d3 = avd[lane + 96];
    for (int i = wid; i < n; i += nw) {
        const float* hr = Hm + (size_t)i * HF;
        float h0 = hr[lane], h1 = hr[lane + 32], h2 = hr[lane + 64], h3 = hr[lane + 96];
        float rs = h0 * s0 + h1 * s1 + h2 * s2 + h3 * s3;
        float rd = h0 * d0 + h1 * d1 + h2 * d2 + h3 * d3;
        for (int off = 16; off; off >>= 1) {
            rs += __shfl_xor(rs, off);
            rd += __shfl_xor(rd, off);
        }
        if (lane == 0) { als[i] = rs; ald[i] = rd; }
    }
}

// self-loop alpha; initializes segment max with the self-loop value
__global__ void k_node_self(const float* __restrict__ als, const float* __restrict__ ald,
                            const float* __restrict__ cvec, float* __restrict__ aself,
                            unsigned* __restrict__ amax_u, int n) {
    int i = blockIdx.x * blockDim.x + threadIdx.x;
    if (i >= n) return;
    float a = leaky(als[i] + ald[i] + cvec[3], 0.2f);
    aself[i] = a;
    amax_u[i] = ordkey(a);
}

// per-edge alpha + atomic segment max
__global__ void k_edge_alpha(const int* __restrict__ src, const int* __restrict__ dst,
                             const float* __restrict__ ea, const float* __restrict__ als,
                             const float* __restrict__ ald, const float* __restrict__ cvec,
                             float* __restrict__ alpha, unsigned* __restrict__ amax_u, int E) {
    int e = blockIdx.x * blockDim.x + threadIdx.x;
    if (e >= E) return;
    float c0 = cvec[0], c1 = cvec[1], c2 = cvec[2];
    float ale = ea[e * 3 + 0] * c0 + ea[e * 3 + 1] * c1 + ea[e * 3 + 2] * c2;
    int s = src[e], d = dst[e];
    float a = leaky(als[s] + ald[d] + ale, 0.2f);
    alpha[e] = a;
    atomicMax(amax_u + d, ordkey(a));
}

// decode max; self-loop exp initializes denom
__global__ void k_node_exself(unsigned* __restrict__ amax_u, float* __restrict__ amaxf,
                              const float* __restrict__ aself, float* __restrict__ exself,
                              float* __restrict__ denom, int n) {
    int i = blockIdx.x * blockDim.x + threadIdx.x;
    if (i >= n) return;
    float m = deord(amax_u[i]);
    float ex = expf(aself[i] - m);
    amaxf[i] = m;        // in-place reuse of the amax buffer as float
    exself[i] = ex;
    denom[i] = ex;
}

// ex = exp(alpha - amax[d]) (in place); atomic segment sum
__global__ void k_edge_exp(float* __restrict__ alpha, const float* __restrict__ amaxf,
                           const int* __restrict__ dst, float* __restrict__ denom, int E) {
    int e = blockIdx.x * blockDim.x + threadIdx.x;
    if (e >= E) return;
    int d = dst[e];
    float ex = expf(alpha[e] - amaxf[d]);
    alpha[e] = ex;
    atomicAdd(denom + d, ex);
}

// out = h * coef_self + bias   (initialize accumulator with self-loop term)
__global__ void k_node_init_out(const float* __restrict__ Hm, const float* __restrict__ exself,
                                const float* __restrict__ denom, const float* __restrict__ bias,
                                float* __restrict__ acc, int n) {
    int idx = blockIdx.x * blockDim.x + threadIdx.x;
    if (idx >= n * HF) return;
    int i = idx >> 7, c = idx & 127;
    float coef = exself[i] / (denom[i] + 1e-16f);
    acc[idx] = Hm[idx] * coef + bias[c];
}

// out[d] += h[s] * coef ; one wave per edge, float4 gather + 4 atomics/lane
__global__ __launch_bounds__(256) void k_edge_scatter(const int* __restrict__ src,
                                                      const int* __restrict__ dst,
                                                      const float* __restrict__ ex,
                                                      const float* __restrict__ denom,
                                                      const float* __restrict__ Hm,
                                                      float* __restrict__ acc, int E) {
    const int lane = threadIdx.x & 31;
    int wid = (blockIdx.x * blockDim.x + threadIdx.x) >> 5;
    int nw  = (gridDim.x * blockDim.x) >> 5;
    for (int e = wid; e < E; e += nw) {
        int s = src[e], d = dst[e];
        float coef = ex[e] / (denom[d] + 1e-16f);
        const float4 hv = *(const float4*)(Hm + (size_t)s * HF + lane * 4);
        float* ap = acc + (size_t)d * HF + lane * 4;
        atomicAdd(ap + 0, hv.x * coef);
        atomicAdd(ap + 1, hv.y * coef);
        atomicAdd(ap + 2, hv.z * coef);
        atomicAdd(ap + 3, hv.w * coef);
    }
}

// BN statistics over leaky(out, 0.01): block of 128 threads (one per feature)
__global__ __launch_bounds__(128) void k_bn_stats(const float* __restrict__ acc,
                                                  float* __restrict__ colsum,
                                                  float* __restrict__ colsq, int n) {
    int c = threadIdx.x;
    int r0 = blockIdx.x * 256;
    int r1 = min(r0 + 256, n);
    float s = 0.f, s2 = 0.f;
    for (int r = r0; r < r1; ++r) {
        float y = leaky(acc[(size_t)r * HF + c], 0.01f);
        s += y; s2 += y * y;
    }
    atomicAdd(colsum + c, s);
    atomicAdd(colsq + c, s2);
}

__global__ void k_bn_apply(const float* __restrict__ acc, const float* __restrict__ colsum,
                           const float* __restrict__ colsq, const float* __restrict__ g,
                           const float* __restrict__ be, float* __restrict__ xout, int n) {
    int idx = blockIdx.x * blockDim.x + threadIdx.x;
    if (idx >= n * HF) return;
    int c = idx & 127;
    float invN = 1.f / (float)n;
    float mu = colsum[c] * invN;
    float var = colsq[c] * invN - mu * mu;
    float y = leaky(acc[idx], 0.01f);
    xout[idx] = (y - mu) * rsqrtf(var + 1e-5f) * g[c] + be[c];
}

// --------------------------------------------------------------------------
extern "C" void kernel_launch(void* const* d_in, const int* in_sizes, int n_in,
                              void* d_out, int out_size, void* d_ws, size_t ws_size,
                              hipStream_t stream) {
    const float* x  = (const float*)d_in[0];
    const int*   ei = (const int*)d_in[1];
    const float* ea = (const float*)d_in[3];
    const int N = in_sizes[0] / HF;
    const int E = in_sizes[1] / 2;
    const int* src = ei;
    const int* dst = ei + E;

    float* ws = (float*)d_ws;
    const size_t NH = (size_t)N * HF;
    float* hbuf   = ws;                 // N*H
    float* acc    = hbuf + NH;          // N*H
    float* xbuf   = acc + NH;           // N*H
    float* alpha  = xbuf + NH;          // E (reused as ex)
    float* als    = alpha + E;          // N
    float* ald    = als + N;            // N
    float* aself  = ald + N;            // N
    float* exself = aself + N;          // N
    float* amaxf  = exself + N;         // N (aliased as unsigned keys first)
    float* denom  = amaxf + N;          // N
    float* colsum = denom + N;          // 128
    float* colsq  = colsum + HF;        // 128
    float* easum  = colsq + HF;         // 4
    float* cvec   = easum + 4;          // 4

    // edge-attr column sums (layer independent)
    k_zero<<<1, 8, 0, stream>>>(easum, 8);
    k_ea_sum<<<128, 256, 0, stream>>>(ea, E, easum);

    const int nb_e  = (E + 255) / 256;
    const int nb_n  = (N + 255) / 256;
    const int nb_nh = (int)((NH + 255) / 256);

    for (int l = 0; l < 3; ++l) {
        const float* Wl  = (const float*)d_in[4 + 8 * l];
        const float* asv = (const float*)d_in[5 + 8 * l];
        const float* adv = (const float*)d_in[6 + 8 * l];
        const float* Wel = (const float*)d_in[7 + 8 * l];
        const float* aev = (const float*)d_in[8 + 8 * l];
        const float* bv  = (const float*)d_in[9 + 8 * l];
        const float* gv  = (const float*)d_in[10 + 8 * l];
        const float* bev = (const float*)d_in[11 + 8 * l];
        const float* xin = (l == 0) ? x : xbuf;
        float* xout = (l == 2) ? (float*)d_out : xbuf;

        k_compute_c<<<1, 1, 0, stream>>>(Wel, aev, easum, cvec, E);
        k_gemm_xw<<<1024, 256, 0, stream>>>(xin, Wl, hbuf, N);
        k_node_dots<<<512, 256, 0, stream>>>(hbuf, asv, adv, als, ald, N);
        k_node_self<<<nb_n, 256, 0, stream>>>(als, ald, cvec, aself, (unsigned*)amaxf, N);
        k_edge_alpha<<<nb_e, 256, 0, stream>>>(src, dst, ea, als, ald, cvec,
                                               alpha, (unsigned*)amaxf, E);
        k_node_exself<<<nb_n, 256, 0, stream>>>((unsigned*)amaxf, amaxf, aself,
                                                exself, denom, N);
        k_edge_exp<<<nb_e, 256, 0, stream>>>(alpha, amaxf, dst, denom, E);
        k_node_init_out<<<nb_nh, 256, 0, stream>>>(hbuf, exself, denom, bv, acc, N);
        k_edge_scatter<<<2048, 256, 0, stream>>>(src, dst, alpha, denom, hbuf, acc, E);
        k_zero<<<1, 256, 0, stream>>>(colsum, 256);  // colsum + colsq (contiguous)
        k_bn_stats<<<(N + 255) / 256, 128, 0, stream>>>(acc, colsum, colsq, N);
        k_bn_apply<<<nb_nh, 256, 0, stream>>>(acc, colsum, colsq, gv, bev, xout, N);
    }
}